// ClusterNet_25494925869839
// MI455X (gfx1250) — compile-verified
//
#include <hip/hip_runtime.h>

typedef float v2f __attribute__((ext_vector_type(2)));
typedef float v8f __attribute__((ext_vector_type(8)));

constexpr int BS = 1 << 20;   // batch rows
constexpr int H  = 64;        // hidden dim (GEMM inner dim)
constexpr int K  = 64;        // clusters (GEMM N)
constexpr int LDSROW = 68;    // padded centroid row stride in floats (bank-conflict free, 16B aligned)
constexpr int TILES_PER_WAVE  = 8;   // 8 x 16 = 128 rows per wave
constexpr int WAVES_PER_BLOCK = 8;   // 256 threads (wave32)
constexpr int ROWS_PER_BLOCK  = 16 * TILES_PER_WAVE * WAVES_PER_BLOCK; // 1024
constexpr int NBLK1 = BS / ROWS_PER_BLOCK;  // 1024 blocks in pass 1

// ---------------------------------------------------------------------------
// Pass 1: WMMA GEMM (z @ c^T) -> d2 -> sim -> Q (row-normalized), plus
// deterministic per-block column-sum partials of Q.
// ---------------------------------------------------------------------------
__global__ __launch_bounds__(256)
void k1_wmma_q(const float* __restrict__ z, const float* __restrict__ cen,
               float* __restrict__ Q, float* __restrict__ gPart)
{
    __shared__ float sC[K * LDSROW];            // centroids, padded rows
    __shared__ float sC2[K];                    // ||c_k||^2
    __shared__ float sPart[WAVES_PER_BLOCK * K];// per-wave column partials

    const int tid  = threadIdx.x;
    const int wave = tid >> 5;
    const int lane = tid & 31;
    const int half = lane >> 4;   // 0: lanes 0-15, 1: lanes 16-31
    const int lrow = lane & 15;

    // Cooperative load of 64x64 centroids into padded LDS (coalesced b32).
    for (int idx = tid; idx < K * H; idx += 256) {
        sC[(idx >> 6) * LDSROW + (idx & 63)] = cen[idx];
    }
    __syncthreads();
    if (tid < K) {
        float s = 0.f;
        #pragma unroll
        for (int h = 0; h < H; ++h) { float v = sC[tid * LDSROW + h]; s += v * v; }
        sC2[tid] = s;
    }
    __syncthreads();

    float colacc[4] = {0.f, 0.f, 0.f, 0.f};
    const v8f vzero = {0.f, 0.f, 0.f, 0.f, 0.f, 0.f, 0.f, 0.f};
    const int waveGlobal = blockIdx.x * WAVES_PER_BLOCK + wave;

    for (int t = 0; t < TILES_PER_WAVE; ++t) {
        const int rowbase = (waveGlobal * TILES_PER_WAVE + t) * 16;

        // A-fragments: 16x4 f32 layout -> lane (half,lrow) holds row `lrow`,
        // K = 4*kk + 2*half + {0,1}. One b64 per lane per K-step; z read once.
        const float* zr = z + (size_t)(rowbase + lrow) * H + 2 * half;
        v2f a[16];
        float zsq = 0.f;
        #pragma unroll
        for (int kk = 0; kk < 16; ++kk) {
            a[kk] = *(const v2f*)(zr + 4 * kk);
            zsq += a[kk].x * a[kk].x + a[kk].y * a[kk].y;
        }
        zsq += __shfl_xor(zsq, 16);   // lane j now holds ||z_{rowbase+(j&15)}||^2

        v8f acc[4];
        #pragma unroll
        for (int nt = 0; nt < 4; ++nt) acc[nt] = vzero;

        // 64 x v_wmma_f32_16x16x4_f32: acc[nt] += A(16x4) * B(4x16)
        #pragma unroll
        for (int nt = 0; nt < 4; ++nt) {
            #pragma unroll
            for (int kk = 0; kk < 16; ++kk) {
                // B[k][n] = centroids[n][k]; lane holds n = 16*nt + lrow,
                // K = 4*kk + 2*half + {0,1}  (8B-aligned ds_load_b64)
                v2f b = *(const v2f*)(&sC[(16 * nt + lrow) * LDSROW + 4 * kk + 2 * half]);
                acc[nt] = __builtin_amdgcn_wmma_f32_16x16x4_f32(
                    false, a[kk], false, b, (short)0, acc[nt], false, false);
            }
        }

        // C/D layout: VGPR v holds row (v + 8*half), column N = lrow + 16*nt.
        #pragma unroll
        for (int v = 0; v < 8; ++v) {
            const float z2 = __shfl(zsq, half ? (v + 8) : v);
            float q[4], rs = 0.f;
            #pragma unroll
            for (int nt = 0; nt < 4; ++nt) {
                const float c2 = sC2[16 * nt + lrow];
                float d2 = z2 + c2 - 2.f * acc[nt][v];
                d2 = fmaxf(d2, 0.f);
                const float sim = __builtin_sqrtf(d2);
                q[nt] = 1.f / (1.f + sim);   // ALPHA==1: (1+sim)^-1
                rs += q[nt];
            }
            // row-sum over 64 columns: 4 tiles done above, 16 lanes per half here
            rs += __shfl_xor(rs, 1);
            rs += __shfl_xor(rs, 2);
            rs += __shfl_xor(rs, 4);
            rs += __shfl_xor(rs, 8);
            const float inv = 1.f / rs;
            const int qrow = rowbase + v + 8 * half;
            #pragma unroll
            for (int nt = 0; nt < 4; ++nt) {
                const float qn = q[nt] * inv;
                colacc[nt] += qn;
                Q[(size_t)qrow * K + 16 * nt + lrow] = qn;
            }
        }
    }

    // Deterministic column-sum partials: wave -> LDS -> block partial.
    #pragma unroll
    for (int nt = 0; nt < 4; ++nt) colacc[nt] += __shfl_xor(colacc[nt], 16);
    if (lane < 16) {
        #pragma unroll
        for (int nt = 0; nt < 4; ++nt) sPart[wave * K + 16 * nt + lane] = colacc[nt];
    }
    __syncthreads();
    if (tid < K) {
        float s = 0.f;
        #pragma unroll
        for (int w = 0; w < WAVES_PER_BLOCK; ++w) s += sPart[w * K + tid];
        gPart[(size_t)blockIdx.x * K + tid] = s;
    }
}

// ---------------------------------------------------------------------------
// Pass 2: reduce NBLK1 block partials -> colsum[64] (fixed order, deterministic)
// ---------------------------------------------------------------------------
__global__ void k2_reduce(const float* __restrict__ gPart, float* __restrict__ colsum)
{
    const int n = threadIdx.x;  // 64 threads
    float s = 0.f;
    for (int i = 0; i < NBLK1; ++i) s += gPart[(size_t)i * K + n];
    colsum[n] = s;
}

// ---------------------------------------------------------------------------
// Pass 3: P = rownorm( Q^2 / colsum ). 16 lanes per row, float4 I/O.
// ---------------------------------------------------------------------------
__global__ __launch_bounds__(256)
void k3_p(const float* __restrict__ Q, const float* __restrict__ colsum,
          float* __restrict__ P)
{
    __shared__ float sF[K];
    if (threadIdx.x < K) sF[threadIdx.x] = colsum[threadIdx.x];
    __syncthreads();

    const int wave = threadIdx.x >> 5;
    const int lane = threadIdx.x & 31;
    const int half = lane >> 4;
    const int j    = lane & 15;

    const int row = (blockIdx.x * WAVES_PER_BLOCK + wave) * 2 + half;
    const float4 q = *(const float4*)(Q + (size_t)row * K + 4 * j);
    const float4 f = *(const float4*)(&sF[4 * j]);
    float4 t;
    t.x = q.x * q.x / f.x;
    t.y = q.y * q.y / f.y;
    t.z = q.z * q.z / f.z;
    t.w = q.w * q.w / f.w;
    float rs = t.x + t.y + t.z + t.w;
    rs += __shfl_xor(rs, 1);
    rs += __shfl_xor(rs, 2);
    rs += __shfl_xor(rs, 4);
    rs += __shfl_xor(rs, 8);
    const float inv = 1.f / rs;
    float4 p;
    p.x = t.x * inv; p.y = t.y * inv; p.z = t.z * inv; p.w = t.w * inv;
    *(float4*)(P + (size_t)row * K + 4 * j) = p;
}

// ---------------------------------------------------------------------------
extern "C" void kernel_launch(void* const* d_in, const int* in_sizes, int n_in,
                              void* d_out, int out_size, void* d_ws, size_t ws_size,
                              hipStream_t stream)
{
    const float* z   = (const float*)d_in[0];   // (BS, H) f32
    const float* cen = (const float*)d_in[1];   // (K, H)  f32

    float* Q = (float*)d_out;                   // (BS, K)
    float* P = Q + (size_t)BS * K;              // (BS, K)

    float* gPart  = (float*)d_ws;               // NBLK1 * K block partials
    float* colsum = gPart + (size_t)NBLK1 * K;  // K column sums

    k1_wmma_q<<<NBLK1, 256, 0, stream>>>(z, cen, Q, gPart);
    k2_reduce<<<1, K, 0, stream>>>(gPart, colsum);
    k3_p<<<BS / (2 * WAVES_PER_BLOCK), 256, 0, stream>>>(Q, colsum, P);
}